// LSNNLayer_26388279067007
// MI455X (gfx1250) — compile-verified
//
#include <hip/hip_runtime.h>

// LSNN recurrent scan for MI455X (gfx1250, wave32).
// T=256 dependent steps; persistent kernel + device-wide barrier; fp32 WMMA.

typedef float v2f __attribute__((ext_vector_type(2)));
typedef float v8f __attribute__((ext_vector_type(8)));

#define T_STEPS 256
#define BSZ     64
#define ISZ     512
#define HSZ     1024

// Model constants (fp32, match reference exactly)
#define DT_TAU_MEM   0.1f        // DT * TAU_MEM_INV
#define DT_TAU_SYN   0.2f        // DT * TAU_SYN_INV
#define DT_TAU_ADAPT 1.25e-6f    // DT * TAU_ADAPT_INV
#define ADAPT_JUMP   2.25e-3f    // TAU_ADAPT_INV * BETA  (note: no DT in reference)
#define V_TH_C       1.0f

#define NUM_BLOCKS 64   // (B/16) * (H/64)

__device__ __forceinline__ void grid_sync(unsigned* cnt, unsigned* gen, unsigned nb) {
    __syncthreads();
    if (threadIdx.x == 0) {
        unsigned g = __hip_atomic_load(gen, __ATOMIC_RELAXED, __HIP_MEMORY_SCOPE_AGENT);
        unsigned prev = __hip_atomic_fetch_add(cnt, 1u, __ATOMIC_ACQ_REL, __HIP_MEMORY_SCOPE_AGENT);
        if (prev == nb - 1u) {
            __hip_atomic_store(cnt, 0u, __ATOMIC_RELAXED, __HIP_MEMORY_SCOPE_AGENT);
            __hip_atomic_store(gen, g + 1u, __ATOMIC_RELEASE, __HIP_MEMORY_SCOPE_AGENT);
        } else {
            while (__hip_atomic_load(gen, __ATOMIC_ACQUIRE, __HIP_MEMORY_SCOPE_AGENT) == g) {
                __builtin_amdgcn_s_sleep(2);
            }
        }
    }
    __syncthreads();
}

// One fp32 WMMA step: D(16x16) += A(16x4) * B(4x16)
__device__ __forceinline__ v8f wmma_f32(v2f a, v2f b, v8f c) {
    return __builtin_amdgcn_wmma_f32_16x16x4_f32(
        /*neg_a=*/false, a, /*neg_b=*/false, b,
        /*c_mod=*/(short)0, c, /*reuse_a=*/false, /*reuse_b=*/false);
}

__global__ void __launch_bounds__(128, 1)
lsnn_persistent(const float* __restrict__ x,    // (T,B,I)
                const float* __restrict__ Wi,   // (H,I)
                const float* __restrict__ Wr,   // (H,H)
                float* __restrict__ out,        // (T,B,H) spikes, then zf,vf,if,bf
                unsigned* __restrict__ bar_cnt,
                unsigned* __restrict__ bar_gen) {
    const int lane  = threadIdx.x & 31;
    const int wv    = threadIdx.x >> 5;        // 4 waves per block
    const int bt    = blockIdx.x & 3;          // batch tile (B/16 = 4)
    const int ht    = blockIdx.x >> 2;         // hidden tile (H/64 = 16)
    const int b0    = bt * 16;
    const int h0    = ht * 64 + wv * 16;       // 16 hidden cols per wave
    const int row   = lane & 15;               // A-matrix M row / B-matrix N col
    const int khalf = lane >> 4;               // 0: K=0,1   1: K=2,3
    const int koff  = khalf * 2;

    // Register-resident per-lane state: C/D layout -> m = khalf*8 + r, n = row
    float vr[8], ir[8], br[8], zr[8];
#pragma unroll
    for (int r = 0; r < 8; ++r) { vr[r] = 0.f; ir[r] = 0.f; br[r] = 0.f; zr[r] = 0.f; }

    const float* WiRow = Wi + (size_t)(h0 + row) * ISZ + koff; // B operand: W[h0+n][k..k+1]
    const float* WrRow = Wr + (size_t)(h0 + row) * HSZ + koff;

    float* zf_tail = out + (size_t)T_STEPS * BSZ * HSZ;

    for (int t = 0; t < T_STEPS; ++t) {
        v8f acc0 = {0.f, 0.f, 0.f, 0.f, 0.f, 0.f, 0.f, 0.f};
        v8f acc1 = {0.f, 0.f, 0.f, 0.f, 0.f, 0.f, 0.f, 0.f};

        // ---- input GEMM: x_t @ W_in^T  (K = ISZ) ----
        const float* xrow = x + ((size_t)t * BSZ + (b0 + row)) * ISZ + koff;
#pragma unroll 4
        for (int k = 0; k < ISZ; k += 8) {
            v2f a0 = *(const v2f*)(xrow + k);
            v2f w0 = *(const v2f*)(WiRow + k);
            acc0 = wmma_f32(a0, w0, acc0);
            v2f a1 = *(const v2f*)(xrow + k + 4);
            v2f w1 = *(const v2f*)(WiRow + k + 4);
            acc1 = wmma_f32(a1, w1, acc1);
        }

        // ---- recurrent GEMM: z_{t-1} @ W_rec^T (K = HSZ); z_{t-1} = outputs[t-1] ----
        if (t > 0) {
            const float* zrow = out + ((size_t)(t - 1) * BSZ + (b0 + row)) * HSZ + koff;
#pragma unroll 4
            for (int k = 0; k < HSZ; k += 8) {
                v2f a0 = *(const v2f*)(zrow + k);
                v2f w0 = *(const v2f*)(WrRow + k);
                acc0 = wmma_f32(a0, w0, acc0);
                v2f a1 = *(const v2f*)(zrow + k + 4);
                v2f w1 = *(const v2f*)(WrRow + k + 4);
                acc1 = wmma_f32(a1, w1, acc1);
            }
        }

        // ---- elementwise LSNN update (state in registers) + spike store ----
        float* zout = out + (size_t)t * BSZ * HSZ;
#pragma unroll
        for (int r = 0; r < 8; ++r) {
            float gemm  = acc0[r] + acc1[r];
            float v_dec = vr[r] + DT_TAU_MEM * (ir[r] - vr[r]);       // V_LEAK = 0
            float b_dec = br[r] + DT_TAU_ADAPT * (V_TH_C - br[r]);
            float z     = (v_dec - b_dec) > 0.f ? 1.f : 0.f;
            vr[r] = (1.f - z) * v_dec;                                 // V_RESET = 0
            br[r] = b_dec + z * ADAPT_JUMP;
            ir[r] = ir[r] * (1.f - DT_TAU_SYN) + gemm;
            zr[r] = z;
            int m = khalf * 8 + r;
            zout[(size_t)(b0 + m) * HSZ + (h0 + row)] = z;
        }

        // make spikes visible to all blocks before next step's recurrent GEMM
        grid_sync(bar_cnt, bar_gen, NUM_BLOCKS);
    }

    // ---- final carry: zf, vf, if, bf ----
#pragma unroll
    for (int r = 0; r < 8; ++r) {
        int m = khalf * 8 + r;
        size_t idx = (size_t)(b0 + m) * HSZ + (h0 + row);
        zf_tail[0 * (size_t)BSZ * HSZ + idx] = zr[r];
        zf_tail[1 * (size_t)BSZ * HSZ + idx] = vr[r];
        zf_tail[2 * (size_t)BSZ * HSZ + idx] = ir[r];
        zf_tail[3 * (size_t)BSZ * HSZ + idx] = br[r];
    }
}

__global__ void lsnn_init(unsigned* bar) {
    if (threadIdx.x < 2) bar[threadIdx.x] = 0u;
}

extern "C" void kernel_launch(void* const* d_in, const int* in_sizes, int n_in,
                              void* d_out, int out_size, void* d_ws, size_t ws_size,
                              hipStream_t stream) {
    const float* x  = (const float*)d_in[0];   // (T,B,I)
    const float* Wi = (const float*)d_in[1];   // (H,I)
    const float* Wr = (const float*)d_in[2];   // (H,H)
    float* out      = (float*)d_out;
    unsigned* bar   = (unsigned*)d_ws;         // [0]=count, [1]=generation

    lsnn_init<<<1, 32, 0, stream>>>(bar);
    lsnn_persistent<<<NUM_BLOCKS, 128, 0, stream>>>(x, Wi, Wr, out, bar, bar + 1);
}